// UNetDecoderBlock_25598005084698
// MI455X (gfx1250) — compile-verified
//
#include <hip/hip_runtime.h>
#include <hip/hip_bf16.h>

typedef float v2f __attribute__((ext_vector_type(2)));
typedef float v8f __attribute__((ext_vector_type(8)));

#define COUT 64
#define CIN  128
#define D2   32
#define DF   64           // full grid dim
#define LDA  68           // LDS row stride (floats); 68*4=272 = 17*16 -> float4-aligned, conflict-free

// ---------------------------------------------------------------------------
// 1) Upsample conv (ConvTranspose3d k=2 s=2) evaluated only at active coords,
//    fused with skip add.
// ---------------------------------------------------------------------------
__global__ __launch_bounds__(64) void k_upconv(const float* __restrict__ x,
                                               const float* __restrict__ skip,
                                               const int*   __restrict__ cords,
                                               const float* __restrict__ Wup,
                                               float* __restrict__ feats, int N) {
  __shared__ float xs[CIN];
  const int n = blockIdx.x;
  const int c = threadIdx.x;
  const int cz = cords[n * 3 + 0], cy = cords[n * 3 + 1], cx = cords[n * 3 + 2];
  const int p   = ((cz & 1) << 2) | ((cy & 1) << 1) | (cx & 1);
  const int off = (cz >> 1) * (D2 * D2) + (cy >> 1) * D2 + (cx >> 1);
  xs[c]      = x[(size_t)c        * (D2 * D2 * D2) + off];
  xs[c + 64] = x[(size_t)(c + 64) * (D2 * D2 * D2) + off];
  __syncthreads();
  float acc = skip[(size_t)c * (DF * DF * DF) + cz * (DF * DF) + cy * DF + cx];
  const float* wp = Wup + (size_t)c * 8 + p;     // Wup[ci, c, p], ci-stride = 512
#pragma unroll 4
  for (int ci = 0; ci < CIN; ++ci)
    acc += xs[ci] * wp[(size_t)ci * (COUT * 8)];
  feats[(size_t)n * COUT + c] = acc;
}

// ---------------------------------------------------------------------------
// 2) coord -> row-index grid, 3) neighbor table nbr[Npad,27] (-1 = masked)
// ---------------------------------------------------------------------------
__global__ void k_grid_init(int* __restrict__ grid, int total) {
  int i = blockIdx.x * blockDim.x + threadIdx.x;
  if (i < total) grid[i] = -1;
}

__global__ void k_scatter(const int* __restrict__ cords, int* __restrict__ grid, int N) {
  int n = blockIdx.x * blockDim.x + threadIdx.x;
  if (n < N)
    grid[cords[n * 3] * (DF * DF) + cords[n * 3 + 1] * DF + cords[n * 3 + 2]] = n;
}

__global__ void k_nbr(const int* __restrict__ cords, const int* __restrict__ grid,
                      int* __restrict__ nbr, int N, int Npad) {
  int i = blockIdx.x * blockDim.x + threadIdx.x;
  if (i >= Npad * 27) return;
  int n = i / 27, o = i % 27;
  int r = -1;
  if (n < N) {
    int az = o / 9 - 1, ay = (o / 3) % 3 - 1, ax = o % 3 - 1;  // OFFS nesting order
    int z = cords[n * 3] + az, y = cords[n * 3 + 1] + ay, x = cords[n * 3 + 2] + ax;
    if ((unsigned)z < (unsigned)DF && (unsigned)y < (unsigned)DF && (unsigned)x < (unsigned)DF)
      r = grid[z * (DF * DF) + y * DF + x];
  }
  nbr[i] = r;
}

// ---------------------------------------------------------------------------
// 4a) Pack W[27,64,64] -> Wt[27][32][64][2]: consecutive K-pair interleaved per
//     column so a B fragment is a single aligned 8-byte load.
// ---------------------------------------------------------------------------
__global__ void k_packW(const float* __restrict__ W, float* __restrict__ Wt) {
  int i = blockIdx.x * blockDim.x + threadIdx.x;
  if (i >= 27 * COUT * COUT) return;
  int o = i / (COUT * COUT);
  int k = (i / COUT) % COUT;
  int n = i % COUT;
  Wt[((size_t)(o * 32 + (k >> 1)) * COUT + n) * 2 + (k & 1)] = W[i];
}

// ---------------------------------------------------------------------------
// 4b) Submanifold conv via V_WMMA_F32_16X16X4_F32.
//     Block = 128 threads (4 waves), 64 rows per block (rows padded; stores
//     unconditional). Wave w owns the 16-row M-tile [n0+16w, n0+16w+16) and
//     computes all 4 N-tiles. A-tile staged in LDS; C staged back through LDS
//     for coalesced b128 stores.
// ---------------------------------------------------------------------------
__global__ __launch_bounds__(128) void k_subm_wmma(const float* __restrict__ fin,
                                                   const int*   __restrict__ nbr,
                                                   const float* __restrict__ Wt,  // packed
                                                   float* __restrict__ fout) {
  __shared__ float As[64 * LDA];
  const int tid  = threadIdx.x;
  const int wave = tid >> 5;
  const int lane = tid & 31;
  const int half = lane >> 4;       // K-half selector (A/B frag layout)
  const int m    = lane & 15;       // M (or N) within tile
  const int n0   = blockIdx.x * 64;

  v8f acc[4];
#pragma unroll
  for (int t = 0; t < 4; ++t) acc[t] = (v8f){0.f, 0.f, 0.f, 0.f, 0.f, 0.f, 0.f, 0.f};

  for (int o = 0; o < 27; ++o) {
    __syncthreads();                 // protect LDS reuse across offsets
    {                                // gather: 2 threads per row, 32 floats each
      const int r   = tid >> 1;
      const int cb  = (tid & 1) * 32;
      const int src = nbr[(size_t)(n0 + r) * 27 + o];
      float4* d4 = (float4*)&As[r * LDA + cb];
      if (src >= 0) {
        const float4* s4 = (const float4*)(fin + (size_t)src * COUT + cb);
#pragma unroll
        for (int q = 0; q < 8; ++q) d4[q] = s4[q];
      } else {
        const float4 z4 = {0.f, 0.f, 0.f, 0.f};
#pragma unroll
        for (int q = 0; q < 8; ++q) d4[q] = z4;
      }
    }
    __syncthreads();

    const float* Arow = &As[(wave * 16 + m) * LDA];
    const v2f*   Bp   = (const v2f*)(Wt + (size_t)o * (32 * COUT * 2));
    for (int k0 = 0; k0 < COUT; k0 += 4) {
      const int kb = k0 + half * 2;        // even
      v2f a;
      a.x = Arow[kb];
      a.y = Arow[kb + 1];
      const v2f* Bk = Bp + (kb >> 1) * COUT;
#pragma unroll
      for (int nt = 0; nt < 4; ++nt) {
        v2f b = Bk[nt * 16 + m];           // one aligned b64 load
        acc[nt] = __builtin_amdgcn_wmma_f32_16x16x4_f32(
            /*neg_a=*/false, a, /*neg_b=*/false, b,
            /*c_mod=*/(short)0, acc[nt], /*reuse_a=*/false, /*reuse_b=*/false);
      }
    }
  }

  // Epilogue: stage C in LDS, then coalesced b128 stores (buffers row-padded).
  __syncthreads();
#pragma unroll
  for (int nt = 0; nt < 4; ++nt)
#pragma unroll
    for (int v = 0; v < 8; ++v)
      As[(wave * 16 + half * 8 + v) * LDA + nt * 16 + m] = acc[nt][v];
  __syncthreads();
  {
    const int r  = tid >> 1;
    const int cb = (tid & 1) * 32;
    const float4* s4 = (const float4*)&As[r * LDA + cb];
    float4* d4 = (float4*)(fout + (size_t)(n0 + r) * COUT + cb);
#pragma unroll
    for (int q = 0; q < 8; ++q) d4[q] = s4[q];
  }
}

// ---------------------------------------------------------------------------
// 5) BatchNorm batch stats (sum / sumsq per channel) + apply
// ---------------------------------------------------------------------------
__global__ void k_zero_stats(float* __restrict__ stats) {
  if (threadIdx.x < 128) stats[threadIdx.x] = 0.f;
}

__global__ __launch_bounds__(64) void k_stats(const float* __restrict__ f,
                                              float* __restrict__ stats, int N) {
  const int c = threadIdx.x;
  float s = 0.f, s2 = 0.f;
  for (int r = blockIdx.x; r < N; r += gridDim.x) {
    float v = f[(size_t)r * COUT + c];
    s += v; s2 += v * v;
  }
  atomicAdd(&stats[c], s);
  atomicAdd(&stats[64 + c], s2);
}

__global__ void k_bn_act(const float* __restrict__ f, const float* __restrict__ stats,
                         const float* __restrict__ g, const float* __restrict__ b,
                         const float* __restrict__ res, float* __restrict__ out,
                         int N, float slope, int addres) {
  int i = blockIdx.x * blockDim.x + threadIdx.x;
  if (i >= N * COUT) return;
  const int c = i & (COUT - 1);
  const float inv  = 1.0f / (float)N;
  const float mean = stats[c] * inv;
  const float var  = stats[64 + c] * inv - mean * mean;
  float t = (f[i] - mean) * (g[c] * rsqrtf(var + 1e-4f)) + b[c];
  if (addres) t += res[i];
  out[i] = (t >= 0.f) ? t : slope * t;
}

// ---------------------------------------------------------------------------
extern "C" void kernel_launch(void* const* d_in, const int* in_sizes, int n_in,
                              void* d_out, int out_size, void* d_ws, size_t ws_size,
                              hipStream_t stream) {
  const float* x     = (const float*)d_in[0];
  const float* skip  = (const float*)d_in[1];
  const int*   cords = (const int*)d_in[2];
  const float* Wup   = (const float*)d_in[3];
  const float* W1    = (const float*)d_in[4];
  const float* W2    = (const float*)d_in[5];
  const float* W3    = (const float*)d_in[6];
  const float* g1 = (const float*)d_in[7],  *b1 = (const float*)d_in[8];
  const float* g2 = (const float*)d_in[9],  *b2 = (const float*)d_in[10];
  const float* g3 = (const float*)d_in[11], *b3 = (const float*)d_in[12];
  float* out = (float*)d_out;

  const int N    = in_sizes[2] / 3;
  const int Npad = (N + 63) & ~63;          // conv kernels write padded rows

  // workspace carve-out (256B aligned slices)
  size_t off = 0;
  auto carve = [&](size_t bytes) -> void* {
    void* p = (char*)d_ws + off;
    off += (bytes + 255) & ~(size_t)255;
    return p;
  };
  float* stats = (float*)carve(128 * sizeof(float));
  int*   grid  = (int*)carve((size_t)DF * DF * DF * sizeof(int));
  int*   nbr   = (int*)carve((size_t)Npad * 27 * sizeof(int));
  float* Wt    = (float*)carve((size_t)27 * COUT * COUT * sizeof(float));
  float* f0    = (float*)carve((size_t)Npad * COUT * sizeof(float));  // upconv out
  float* raw   = (float*)carve((size_t)Npad * COUT * sizeof(float));  // conv raw out
  float* h1    = (float*)carve((size_t)Npad * COUT * sizeof(float));  // residual
  float* h2    = (float*)carve((size_t)Npad * COUT * sizeof(float));

  const int gridTotal = DF * DF * DF;

  // stage 0: features at active voxels + neighbor table
  k_upconv<<<N, 64, 0, stream>>>(x, skip, cords, Wup, f0, N);
  k_grid_init<<<(gridTotal + 255) / 256, 256, 0, stream>>>(grid, gridTotal);
  k_scatter<<<(N + 255) / 256, 256, 0, stream>>>(cords, grid, N);
  k_nbr<<<(Npad * 27 + 255) / 256, 256, 0, stream>>>(cords, grid, nbr, N, Npad);

  const int convBlocks = Npad / 64;
  const int elemBlocks = (N * COUT + 255) / 256;
  const int packBlocks = (27 * COUT * COUT + 255) / 256;

  // stage 1: h1 = leaky(bn(conv(f0,W1)), 0.05)
  k_packW<<<packBlocks, 256, 0, stream>>>(W1, Wt);
  k_subm_wmma<<<convBlocks, 128, 0, stream>>>(f0, nbr, Wt, raw);
  k_zero_stats<<<1, 128, 0, stream>>>(stats);
  k_stats<<<512, 64, 0, stream>>>(raw, stats, N);
  k_bn_act<<<elemBlocks, 256, 0, stream>>>(raw, stats, g1, b1, nullptr, h1, N, 0.05f, 0);

  // stage 2: h2 = leaky(bn(conv(h1,W2)), 0.05)
  k_packW<<<packBlocks, 256, 0, stream>>>(W2, Wt);
  k_subm_wmma<<<convBlocks, 128, 0, stream>>>(h1, nbr, Wt, raw);
  k_zero_stats<<<1, 128, 0, stream>>>(stats);
  k_stats<<<512, 64, 0, stream>>>(raw, stats, N);
  k_bn_act<<<elemBlocks, 256, 0, stream>>>(raw, stats, g2, b2, nullptr, h2, N, 0.05f, 0);

  // stage 3: out = leaky(bn(conv(h2,W3)) + h1, 1/3)
  k_packW<<<packBlocks, 256, 0, stream>>>(W3, Wt);
  k_subm_wmma<<<convBlocks, 128, 0, stream>>>(h2, nbr, Wt, raw);
  k_zero_stats<<<1, 128, 0, stream>>>(stats);
  k_stats<<<512, 64, 0, stream>>>(raw, stats, N);
  k_bn_act<<<elemBlocks, 256, 0, stream>>>(raw, stats, g3, b3, h1, out, N,
                                           1.0f / 3.0f, 1);
}